// FeatureExtracter_57071525430150
// MI455X (gfx1250) — compile-verified
//
#include <hip/hip_runtime.h>
#include <hip/hip_bf16.h>

typedef float v2f __attribute__((ext_vector_type(2)));
typedef float v8f __attribute__((ext_vector_type(8)));

// ---------------------------------------------------------------------------
// Utility: zero a float buffer
// ---------------------------------------------------------------------------
__global__ void zero_kernel(float* __restrict__ p, long n) {
    long gid = (long)blockIdx.x * blockDim.x + threadIdx.x;
    if (gid < n) p[gid] = 0.0f;
}

// ---------------------------------------------------------------------------
// Degree counts: outdeg[src[e]] += 1, indeg[dst[e]] += 1 (float atomics)
// ---------------------------------------------------------------------------
__global__ void degree_kernel(const int* __restrict__ src,
                              const int* __restrict__ dst,
                              long E,
                              float* __restrict__ outdeg,
                              float* __restrict__ indeg) {
    long gid = (long)blockIdx.x * blockDim.x + threadIdx.x;
    if (gid >= E) return;
    atomicAdd(&outdeg[src[gid]], 1.0f);
    atomicAdd(&indeg[dst[gid]], 1.0f);
}

// ---------------------------------------------------------------------------
// deg -> rsqrt(max(deg, 1)) in place
// ---------------------------------------------------------------------------
__global__ void rsqrt_kernel(float* __restrict__ deg, long n) {
    long gid = (long)blockIdx.x * blockDim.x + threadIdx.x;
    if (gid < n) deg[gid] = rsqrtf(fmaxf(deg[gid], 1.0f));
}

// ---------------------------------------------------------------------------
// H = (X @ W) * norm_row[:, None]   via V_WMMA_F32_16X16X4_F32
// X: [N, d] row-major, W: [d, d] row-major, H: [N, d].
// One wave computes one 16x16 output tile. blockDim.x = 256 (8 waves).
// N % 16 == 0 and total_tiles % 8 == 0 for our sizes, so EXEC is all-1s
// for every wave that runs a WMMA (required by ISA).
//
// VGPR layouts (CDNA5 ISA 7.12.2, 32-bit data, wave32):
//   A 16x4 : lane<16 -> M=lane, {a.x,a.y}={K0,K1}; lane>=16 -> M=lane-16, {K2,K3}
//   B 4x16 : VGPR v, lanes 0-15 hold row K=v (N=lane), lanes 16-31 row K=v+2
//   C/D    : VGPR i, lane l -> (M = i + 8*(l>=16), N = l&15)
// ---------------------------------------------------------------------------
__global__ void gemm_norm_kernel(const float* __restrict__ X,
                                 const float* __restrict__ W,
                                 const float* __restrict__ norm,
                                 float* __restrict__ H,
                                 int N, int d) {
    const int lane    = threadIdx.x & 31;
    const int wave    = threadIdx.x >> 5;
    const int tiles_n = d >> 4;                       // column tiles
    const long tile   = (long)blockIdx.x * (blockDim.x >> 5) + wave;
    const long total  = (long)(N >> 4) * tiles_n;
    if (tile >= total) return;                        // wave-uniform

    const int tm      = (int)(tile / tiles_n);
    const int tn      = (int)(tile % tiles_n);
    const int l15     = lane & 15;
    const int half    = lane >> 4;                    // 0 or 1
    const int kb      = half * 2;                     // K sub-offset for this half-wave
    const int row     = tm * 16 + l15;                // A row for this lane
    const int col     = tn * 16 + l15;                // B column for this lane

    const float* __restrict__ Xrow = X + (size_t)row * d;

    v8f acc = {0.f, 0.f, 0.f, 0.f, 0.f, 0.f, 0.f, 0.f};

    for (int k = 0; k < d; k += 4) {
        v2f a;
        a.x = Xrow[k + kb];
        a.y = Xrow[k + kb + 1];
        v2f b;
        b.x = W[(size_t)(k + kb) * d + col];
        b.y = W[(size_t)(k + kb + 1) * d + col];
        // D = A(16x4) x B(4x16) + C
        acc = __builtin_amdgcn_wmma_f32_16x16x4_f32(
            /*neg_a=*/false, a, /*neg_b=*/false, b,
            /*c_mod=*/(short)0, acc, /*reuse_a=*/false, /*reuse_b=*/false);
    }

    // Epilogue: scale each output row by norm_src[row] and store.
#pragma unroll
    for (int i = 0; i < 8; ++i) {
        const int m = tm * 16 + i + half * 8;
        H[(size_t)m * d + tn * 16 + l15] = acc[i] * norm[m];
    }
}

// ---------------------------------------------------------------------------
// Edge scatter: out[dst[e]] += H[src[e]]  (H already scaled by norm_src)
// One thread handles 4 consecutive floats of one edge (float4 gather,
// 4x GLOBAL_ATOMIC_ADD_F32 without return). chunk_shift = log2(d/4).
// ---------------------------------------------------------------------------
__global__ void scatter_kernel(const float* __restrict__ H,
                               const int* __restrict__ src,
                               const int* __restrict__ dst,
                               float* __restrict__ out,
                               long E, int d, int chunk_shift) {
    const long chunks = 1L << chunk_shift;
    const long gid    = (long)blockIdx.x * blockDim.x + threadIdx.x;
    if (gid >= E * chunks) return;
    const long e = gid >> chunk_shift;
    const int  c = (int)(gid & (chunks - 1));

    const int s = src[e];
    const int t = dst[e];

    const float4 v = ((const float4*)(H + (size_t)s * d))[c];
    float* o = out + (size_t)t * d + (size_t)c * 4;
    atomicAdd(o + 0, v.x);
    atomicAdd(o + 1, v.y);
    atomicAdd(o + 2, v.z);
    atomicAdd(o + 3, v.w);
}

// ---------------------------------------------------------------------------
// Finalize in place: out = relu(out * norm_dst[n] + bias[f])
// ---------------------------------------------------------------------------
__global__ void finalize_kernel(float* __restrict__ out,
                                const float* __restrict__ norm,
                                const float* __restrict__ bias,
                                long N, int d, int chunk_shift) {
    const long chunks = 1L << chunk_shift;
    const long gid    = (long)blockIdx.x * blockDim.x + threadIdx.x;
    if (gid >= N * chunks) return;
    const long n = gid >> chunk_shift;
    const int  c = (int)(gid & (chunks - 1));
    const float s = norm[n];

    float4* p = (float4*)(out + (size_t)n * d) + c;
    float4 v = *p;
    const float4 b = ((const float4*)bias)[c];
    v.x = fmaxf(v.x * s + b.x, 0.0f);
    v.y = fmaxf(v.y * s + b.y, 0.0f);
    v.z = fmaxf(v.z * s + b.z, 0.0f);
    v.w = fmaxf(v.w * s + b.w, 0.0f);
    *p = v;
}

// ---------------------------------------------------------------------------
// Launch
// ---------------------------------------------------------------------------
extern "C" void kernel_launch(void* const* d_in, const int* in_sizes, int n_in,
                              void* d_out, int out_size, void* d_ws, size_t ws_size,
                              hipStream_t stream) {
    const float* feat_c = (const float*)d_in[0];
    const float* feat_d = (const float*)d_in[1];
    const int*   src_c  = (const int*)d_in[2];
    const int*   dst_c  = (const int*)d_in[3];
    const int*   src_d  = (const int*)d_in[4];
    const int*   dst_d  = (const int*)d_in[5];
    const float* W_c    = (const float*)d_in[6];
    const float* b_c    = (const float*)d_in[7];
    const float* W_d    = (const float*)d_in[8];
    const float* b_d    = (const float*)d_in[9];

    const int  dc = 128, dd = 256;
    const long N  = (long)in_sizes[0] / dc;     // 50000
    const long Ec = (long)in_sizes[2];          // 1.6M
    const long Ed = (long)in_sizes[4];          // 0.8M

    float* out   = (float*)d_out;
    float* out_c = out;
    float* out_d = out + N * dc;

    // Workspace layout (floats): h_c | h_d | outdeg_c | indeg_c | outdeg_d | indeg_d
    float* ws       = (float*)d_ws;
    float* h_c      = ws;
    float* h_d      = h_c + N * dc;
    float* outdeg_c = h_d + N * dd;
    float* indeg_c  = outdeg_c + N;
    float* outdeg_d = indeg_c + N;
    float* indeg_d  = outdeg_d + N;

    const int B = 256;
    auto blocks = [](long n, int b) { return (unsigned)((n + b - 1) / b); };

    // 1) Zero accumulators (d_out) and degree arrays (deterministic per launch).
    zero_kernel<<<blocks((long)out_size, B), B, 0, stream>>>(out, (long)out_size);
    zero_kernel<<<blocks(4 * N, B), B, 0, stream>>>(outdeg_c, 4 * N);

    // 2) Degrees.
    degree_kernel<<<blocks(Ec, B), B, 0, stream>>>(src_c, dst_c, Ec, outdeg_c, indeg_c);
    degree_kernel<<<blocks(Ed, B), B, 0, stream>>>(src_d, dst_d, Ed, outdeg_d, indeg_d);

    // 3) norm = rsqrt(clamp(deg, 1)) in place for all four arrays.
    rsqrt_kernel<<<blocks(4 * N, B), B, 0, stream>>>(outdeg_c, 4 * N);

    // 4) H = (X @ W) * norm_src (fp32 WMMA, one 16x16 tile per wave, 8 waves/block).
    {
        long tiles_c = (N >> 4) * (dc >> 4);    // 25000, divisible by 8
        long tiles_d = (N >> 4) * (dd >> 4);    // 50000, divisible by 8
        gemm_norm_kernel<<<blocks(tiles_c, 8), B, 0, stream>>>(feat_c, W_c, outdeg_c, h_c, (int)N, dc);
        gemm_norm_kernel<<<blocks(tiles_d, 8), B, 0, stream>>>(feat_d, W_d, outdeg_d, h_d, (int)N, dd);
    }

    // 5) Edge scatter-add directly into d_out (L2-resident atomics).
    scatter_kernel<<<blocks(Ec * (dc / 4), B), B, 0, stream>>>(h_c, src_c, dst_c, out_c, Ec, dc, 5);
    scatter_kernel<<<blocks(Ed * (dd / 4), B), B, 0, stream>>>(h_d, src_d, dst_d, out_d, Ed, dd, 6);

    // 6) Finalize: relu(agg * norm_dst + bias), in place.
    finalize_kernel<<<blocks(N * (dc / 4), B), B, 0, stream>>>(out_c, indeg_c, b_c, N, dc, 5);
    finalize_kernel<<<blocks(N * (dd / 4), B), B, 0, stream>>>(out_d, indeg_d, b_d, N, dd, 6);
}